// NAGSL_21105469293032
// MI455X (gfx1250) — compile-verified
//
#include <hip/hip_runtime.h>
#include <math.h>

// ---------------------------------------------------------------------------
// Types for WMMA fragments (fp32, 16x16x4)
// ---------------------------------------------------------------------------
typedef __attribute__((ext_vector_type(2))) float v2f;
typedef __attribute__((ext_vector_type(8))) float v8f;

#ifndef __has_builtin
#define __has_builtin(x) 0
#endif
#define HAVE_WMMA_F32X4 __has_builtin(__builtin_amdgcn_wmma_f32_16x16x4_f32)

// activation codes: 0 none, 1 relu, 2 exact gelu, 3 sigmoid, 4 leaky relu(0.33)
__device__ __forceinline__ float apply_act(float x, int act) {
  switch (act) {
    case 1: return x > 0.f ? x : 0.f;
    case 2: return 0.5f * x * (1.0f + erff(x * 0.70710678118654752f));
    case 3: return 1.0f / (1.0f + __expf(-x));
    case 4: return x >= 0.f ? x : 0.33f * x;
    default: return x;
  }
}

// ---------------------------------------------------------------------------
// GCN layer (PyG GCNConv semantics + relu). One workgroup, n<=64, feat=64.
// ---------------------------------------------------------------------------
__global__ void k_gcn(const float* __restrict__ x, const float* __restrict__ w,
                      const float* __restrict__ b, const int* __restrict__ src,
                      const int* __restrict__ dst, int E, int n,
                      float* __restrict__ out) {
  __shared__ float sh_h[64 * 64];
  __shared__ float sh_out[64 * 64];
  __shared__ float sh_deg[64];
  int tid = threadIdx.x;  // 256 threads
  for (int i = tid; i < 64; i += 256) sh_deg[i] = (i < n) ? 1.0f : 0.0f;
  __syncthreads();
  for (int e = tid; e < E; e += 256) atomicAdd(&sh_deg[dst[e]], 1.0f);
  // h = x @ w^T
  for (int idx = tid; idx < n * 64; idx += 256) {
    int r = idx >> 6, c = idx & 63;
    float s = 0.f;
    for (int k = 0; k < 64; ++k) s += x[r * 64 + k] * w[c * 64 + k];
    sh_h[idx] = s;
  }
  __syncthreads();  // deg + h complete
  for (int idx = tid; idx < n * 64; idx += 256) {
    int r = idx >> 6;
    sh_out[idx] = sh_h[idx] / sh_deg[r];  // self loop: dinv*dinv = 1/deg
  }
  __syncthreads();
  for (int idx = tid; idx < E * 64; idx += 256) {
    int e = idx >> 6, c = idx & 63;
    int s = src[e], d = dst[e];
    float nrm = rsqrtf(sh_deg[s]) * rsqrtf(sh_deg[d]);
    atomicAdd(&sh_out[d * 64 + c], nrm * sh_h[s * 64 + c]);
  }
  __syncthreads();
  for (int idx = tid; idx < n * 64; idx += 256) {
    int c = idx & 63;
    float v = sh_out[idx] + b[c];
    out[idx] = v > 0.f ? v : 0.f;
  }
}

__global__ void k_add3_pad(const float* a, const float* b, const float* c,
                           int n, float* g) {
  int i = blockIdx.x * blockDim.x + threadIdx.x;
  if (i >= 64 * 64) return;
  int r = i >> 6;
  g[i] = (r < n) ? (a[i] + b[i] + c[i]) : 0.f;
}

__global__ void k_add(const float* a, const float* b, float* o, int n) {
  int i = blockIdx.x * blockDim.x + threadIdx.x;
  if (i < n) o[i] = a[i] + b[i];
}

// LayerNorm over rows; one block (256 thr) per row.
__global__ void k_ln(const float* __restrict__ in, const float* __restrict__ g,
                     const float* __restrict__ b, float* __restrict__ out, int C) {
  __shared__ float red[256];
  int row = blockIdx.x, tid = threadIdx.x;
  const float* xr = in + (size_t)row * C;
  float s = 0.f;
  for (int i = tid; i < C; i += 256) s += xr[i];
  red[tid] = s;
  __syncthreads();
  for (int st = 128; st > 0; st >>= 1) {
    if (tid < st) red[tid] += red[tid + st];
    __syncthreads();
  }
  float mean = red[0] / (float)C;
  __syncthreads();
  float v = 0.f;
  for (int i = tid; i < C; i += 256) {
    float d = xr[i] - mean;
    v += d * d;
  }
  red[tid] = v;
  __syncthreads();
  for (int st = 128; st > 0; st >>= 1) {
    if (tid < st) red[tid] += red[tid + st];
    __syncthreads();
  }
  float rstd = rsqrtf(red[0] / (float)C + 1e-5f);
  for (int i = tid; i < C; i += 256)
    out[(size_t)row * C + i] = (xr[i] - mean) * rstd * g[i] + b[i];
}

// Generic small GEMM: out[M,N] = act(A[M,K] @ W[N,K]^T + bias [+ res])
__global__ void k_gemm_rt(const float* __restrict__ A, const float* __restrict__ W,
                          const float* __restrict__ bias, const float* __restrict__ res,
                          float* __restrict__ out, int M, int K, int N, int act) {
  int i = blockIdx.x * blockDim.x + threadIdx.x;
  if (i >= M * N) return;
  int m = i / N, n = i % N;
  const float* a = A + (size_t)m * K;
  const float* w = W + (size_t)n * K;
  float s = 0.f;
  for (int k = 0; k < K; ++k) s += a[k] * w[k];
  s += bias[n];
  if (res) s += res[i];
  out[i] = apply_act(s, act);
}

// Encoder self-attention with torch's view(H,-1,D) row-mixing reshape.
// q[h,n,d] = qb[(h*8 + n/8)*512 + (n%8)*64 + d];  SCALE = 1/8.
__global__ void k_attn_enc(const float* __restrict__ qb, const float* __restrict__ kb,
                           const float* __restrict__ vb, float* __restrict__ yb) {
  __shared__ float sc[64];
  __shared__ float red[64];
  int h = blockIdx.x >> 6, n = blockIdx.x & 63;
  int t = threadIdx.x;  // 64 threads
  auto eidx = [](int hh, int nn, int dd) {
    return (hh * 8 + (nn >> 3)) * 512 + ((nn & 7) << 6) + dd;
  };
  float s = 0.f;
  for (int d = 0; d < 64; ++d) s += qb[eidx(h, n, d)] * kb[eidx(h, t, d)];
  s *= 0.125f;
  sc[t] = s;
  red[t] = s;
  __syncthreads();
  for (int st = 32; st > 0; st >>= 1) {
    if (t < st) red[t] = fmaxf(red[t], red[t + st]);
    __syncthreads();
  }
  float mx = red[0];
  __syncthreads();
  float e = __expf(sc[t] - mx);
  sc[t] = e;
  red[t] = e;
  __syncthreads();
  for (int st = 32; st > 0; st >>= 1) {
    if (t < st) red[t] += red[t + st];
    __syncthreads();
  }
  float inv = 1.0f / red[0];
  __syncthreads();
  float y = 0.f;
  for (int m = 0; m < 64; ++m) y += sc[m] * vb[eidx(h, m, t)];
  yb[n * 512 + h * 64 + t] = y * inv;
}

// Cross attention scores -> sim tensor t[16, 64, 64] (standard per-head split)
__global__ void k_cross(const float* __restrict__ qi, const float* __restrict__ ki,
                        const float* __restrict__ qj, const float* __restrict__ kj,
                        float* __restrict__ t) {
  int i = blockIdx.x * blockDim.x + threadIdx.x;
  if (i >= 16 * 4096) return;
  int c = i >> 12, rem = i & 4095, n = rem >> 6, m = rem & 63;
  auto idx = [](int h, int nn, int d) { return nn * 512 + h * 64 + d; };
  float s = 0.f;
  if (c < 8) {
    for (int d = 0; d < 64; ++d) s += qi[idx(c, n, d)] * kj[idx(c, m, d)];
  } else {
    int h = c - 8;
    for (int d = 0; d < 64; ++d) s += qj[idx(h, m, d)] * ki[idx(h, n, d)];
  }
  t[i] = s * 0.125f;
}

// ---------------------------------------------------------------------------
// Big skinny GEMMs: partial[ks] = A[16, kchunk] @ W[.,kchunk]^T using
// V_WMMA_F32_16X16X4_F32, K split across blockIdx.y for memory parallelism.
// Software pipeline with two REGISTER-STATIC fragment sets (no dynamic
// indexing -> no v_movrel): loads for batch b+2 issue before the WMMAs that
// consume batch b+1, keeping ~12 b64 loads in flight per wave.
// Fragment packing per ISA 7.12.2 (32-bit A 16x4 / C,D 16x16 layouts).
// ---------------------------------------------------------------------------
#define GW_TILES 2
#define GW_UNROLL 4
#define GW_KB (4 * GW_UNROLL)  // k advanced per batch

struct GwFrag {
  v2f a[GW_UNROLL];
  v2f b[GW_TILES][GW_UNROLL];
};

__device__ __forceinline__ void gw_load(GwFrag& f, const float* __restrict__ arow,
                                        const float* const* wrow, int kbase) {
#pragma unroll
  for (int u = 0; u < GW_UNROLL; ++u) {
    f.a[u].x = arow[kbase + 4 * u];
    f.a[u].y = arow[kbase + 4 * u + 1];
  }
#pragma unroll
  for (int t = 0; t < GW_TILES; ++t)
#pragma unroll
    for (int u = 0; u < GW_UNROLL; ++u) {
      f.b[t][u].x = wrow[t][kbase + 4 * u];
      f.b[t][u].y = wrow[t][kbase + 4 * u + 1];
    }
}

__device__ __forceinline__ void gw_compute(const GwFrag& f, v8f* acc) {
#if HAVE_WMMA_F32X4
#pragma unroll
  for (int u = 0; u < GW_UNROLL; ++u)
#pragma unroll
    for (int t = 0; t < GW_TILES; ++t)
      acc[t] = __builtin_amdgcn_wmma_f32_16x16x4_f32(
          false, f.a[u], false, f.b[t][u], (short)0, acc[t], false, false);
#else
  (void)f; (void)acc;
#endif
}

__global__ void __launch_bounds__(128)
k_gemm_wmma_part(const float* __restrict__ A, const float* __restrict__ W,
                 float* __restrict__ part, int K, int N, int kchunk) {
#if HAVE_WMMA_F32X4
  int wave = threadIdx.x >> 5;
  int lane = threadIdx.x & 31;
  int nb = (blockIdx.x * (blockDim.x >> 5) + wave) * (16 * GW_TILES);
  int l15 = lane & 15;
  int koff = (lane >> 4) << 1;  // lanes 0-15 -> K{0,1}, lanes 16-31 -> K{2,3}
  int k0 = blockIdx.y * kchunk;
  const float* arow = A + (size_t)l15 * K + k0 + koff;
  const float* wrow[GW_TILES];
#pragma unroll
  for (int t = 0; t < GW_TILES; ++t)
    wrow[t] = W + (size_t)(nb + t * 16 + l15) * K + k0 + koff;

  const v8f vzero = {0.f, 0.f, 0.f, 0.f, 0.f, 0.f, 0.f, 0.f};
  v8f acc[GW_TILES];
#pragma unroll
  for (int t = 0; t < GW_TILES; ++t) acc[t] = vzero;

  // nbatch = kchunk/16 is 32 or 64 here: even and >= 2.
  int nbatch = kchunk / GW_KB;
  GwFrag f0, f1;
  gw_load(f0, arow, wrow, 0);
  gw_load(f1, arow, wrow, GW_KB);
  int kbase = 2 * GW_KB;
  for (int i = 0; i < (nbatch >> 1) - 1; ++i) {
    gw_compute(f0, acc);
    gw_load(f0, arow, wrow, kbase);
    gw_compute(f1, acc);
    gw_load(f1, arow, wrow, kbase + GW_KB);
    kbase += 2 * GW_KB;
  }
  gw_compute(f0, acc);
  gw_compute(f1, acc);

  int mb = (lane < 16) ? 0 : 8;
  float* pp = part + (size_t)blockIdx.y * 16 * N;
#pragma unroll
  for (int t = 0; t < GW_TILES; ++t) {
    int n = nb + t * 16 + l15;
#pragma unroll
    for (int r = 0; r < 8; ++r) pp[(size_t)(mb + r) * N + n] = acc[t][r];
  }
#else
  // Scalar fallback (also what the host pass sees)
  int total = 16 * N;
  int k0 = blockIdx.y * kchunk;
  float* pp = part + (size_t)blockIdx.y * 16 * N;
  for (int i = blockIdx.x * blockDim.x + threadIdx.x; i < total;
       i += gridDim.x * blockDim.x) {
    int m = i / N, n = i % N;
    const float* a = A + (size_t)m * K + k0;
    const float* w = W + (size_t)n * K + k0;
    float s = 0.f;
    for (int k = 0; k < kchunk; ++k) s += a[k] * w[k];
    pp[i] = s;
  }
#endif
}

// Deterministic fixed-order reduction of K-split partials + bias/res/act.
__global__ void k_gemm_finalize(const float* __restrict__ part, int nparts,
                                const float* __restrict__ bias,
                                const float* __restrict__ res,
                                float* __restrict__ out, int N, int act) {
  int i = blockIdx.x * blockDim.x + threadIdx.x;
  if (i >= 16 * N) return;
  float s = 0.f;
  for (int p = 0; p < nparts; ++p) s += part[(size_t)p * 16 * N + i];
  s += bias[i % N];
  if (res) s += res[i];
  out[i] = apply_act(s, act);
}

// channel-align attention: scores[h,n,m] = (1/64) * dot(q[n,h*4096:], k[m,h*4096:])
__global__ void k_ca_scores(const float* __restrict__ q, const float* __restrict__ k,
                            float* __restrict__ sc) {
  __shared__ float red[128];
  int b = blockIdx.x;  // h*256 + n*16 + m
  int h = b >> 8, n = (b >> 4) & 15, m = b & 15;
  const float* qp = q + (size_t)n * 8192 + h * 4096;
  const float* kp = k + (size_t)m * 8192 + h * 4096;
  float s = 0.f;
  for (int d = threadIdx.x; d < 4096; d += 128) s += qp[d] * kp[d];
  red[threadIdx.x] = s;
  __syncthreads();
  for (int st = 64; st > 0; st >>= 1) {
    if (threadIdx.x < st) red[threadIdx.x] += red[threadIdx.x + st];
    __syncthreads();
  }
  if (threadIdx.x == 0) sc[b] = red[0] * (1.0f / 64.0f);  // CSCALE = 4096^-0.5
}

__global__ void k_ca_softmax(const float* __restrict__ sc, float* __restrict__ a) {
  int i = threadIdx.x;  // 32 = (h,n) pairs
  if (i >= 32) return;
  const float* s = sc + i * 16;
  float mx = -1e30f;
  for (int m = 0; m < 16; ++m) mx = fmaxf(mx, s[m]);
  float e[16], sum = 0.f;
  for (int m = 0; m < 16; ++m) {
    e[m] = __expf(s[m] - mx);
    sum += e[m];
  }
  float inv = 1.0f / sum;
  for (int m = 0; m < 16; ++m) a[i * 16 + m] = e[m] * inv;
}

__global__ void k_ca_av(const float* __restrict__ a, const float* __restrict__ v,
                        float* __restrict__ y) {
  int i = blockIdx.x * blockDim.x + threadIdx.x;  // 16*8192
  if (i >= 16 * 8192) return;
  int n = i >> 13, c = i & 8191;
  int h = c >> 12;
  const float* ar = a + (h * 16 + n) * 16;
  float s = 0.f;
  for (int m = 0; m < 16; ++m) s += ar[m] * v[(size_t)m * 8192 + c];
  y[i] = s;
}

// E2E-style conv projections. mode 0: out[o,h]=sum_{c,w} x[c,h,w]*W[o,c,w]
//                             mode 1: out[o,w]=sum_{c,h} x[c,h,w]*W[o,c,h]
__global__ void k_e2e_proj(const float* __restrict__ x, const float* __restrict__ w,
                           const float* __restrict__ bias, float* __restrict__ out,
                           int Cin, int Cout, int mode, int act) {
  int i = blockIdx.x * blockDim.x + threadIdx.x;
  if (i >= Cout * 64) return;
  int o = i >> 6, p = i & 63;
  float s = 0.f;
  for (int c = 0; c < Cin; ++c) {
    const float* wp = w + ((size_t)o * Cin + c) * 64;
    const float* xp = x + (size_t)c * 4096;
    if (mode == 0) {
      const float* xr = xp + p * 64;
      for (int q = 0; q < 64; ++q) s += xr[q] * wp[q];
    } else {
      for (int q = 0; q < 64; ++q) s += xp[q * 64 + p] * wp[q];
    }
  }
  s += bias[o];
  out[i] = apply_act(s, act);
}

__global__ void k_e2e_combine(const float* __restrict__ A, const float* __restrict__ B,
                              float* __restrict__ out, int Cout) {
  int i = blockIdx.x * blockDim.x + threadIdx.x;
  if (i >= Cout * 4096) return;
  int o = i >> 12, hw = i & 4095, h = hw >> 6, w = hw & 63;
  float v = A[o * 64 + h] + B[o * 64 + w];
  out[i] = v >= 0.f ? v : 0.33f * v;
}

// ---------------------------------------------------------------------------
// Host side
// ---------------------------------------------------------------------------
enum {
  IN_XI = 0, IN_XJ, IN_EI, IN_EJ,
  P_caf1_b, P_caf1_w, P_caf2_b, P_caf2_w, P_cafn_b, P_cafn_g,
  P_cak_b, P_cak_w, P_cao_b, P_cao_w, P_caq_b, P_caq_w, P_cav_b, P_cav_w,
  P_ck_b, P_ck_w, P_cq_b, P_cq_w,
  P_e2e1a_b, P_e2e1a_w, P_e2e1b_b, P_e2e1b_w,
  P_e2e2a_b, P_e2e2a_w, P_e2e2b_b, P_e2e2b_w,
  P_e2n_b, P_e2n_w, P_f1_b, P_f1_w, P_f2_b, P_f2_w,
  P_fc1_b, P_fc1_w, P_fc2_b, P_fc2_w, P_fc3_b, P_fc3_w, P_fc4_b, P_fc4_w,
  P_fn_b, P_fn_g, P_gcn1_b, P_gcn1_w, P_gcn2_b, P_gcn2_w, P_gcn3_b, P_gcn3_w,
  P_k_b, P_k_w, P_n2g_b, P_n2g_w, P_o_b, P_o_w, P_q_b, P_q_w,
  P_san_b, P_san_g, P_v_b, P_v_w
};

extern "C" void kernel_launch(void* const* d_in, const int* in_sizes, int n_in,
                              void* d_out, int out_size, void* d_ws, size_t ws_size,
                              hipStream_t stream) {
  (void)n_in; (void)out_size; (void)ws_size;
  auto P = [&](int i) { return (const float*)d_in[i]; };
  const float* xi = (const float*)d_in[IN_XI];
  const float* xj = (const float*)d_in[IN_XJ];
  const int* epi = (const int*)d_in[IN_EI];
  const int* epj = (const int*)d_in[IN_EJ];
  int n1 = in_sizes[IN_XI] / 64, n2 = in_sizes[IN_XJ] / 64;
  int E1 = in_sizes[IN_EI] / 2, E2 = in_sizes[IN_EJ] / 2;

  float* ws = (float*)d_ws;
  float* h1 = ws + 0;          // 4096
  float* h2 = ws + 4096;       // 4096
  float* h3 = ws + 8192;       // 4096
  float* g  = ws + 12288;      // 4096
  float* yln = ws + 16384;     // 4096
  float* qb = ws + 20480;      // 32768
  float* kb = ws + 53248;      // 32768
  float* vb = ws + 86016;      // 32768
  float* yb = ws + 118784;     // 32768
  float* sbuf = ws + 151552;   // 4096
  float* fbuf = ws + 155648;   // 4096
  float* ffn1 = ws + 159744;   // 16384
  float* s2 = ws + 176128;     // 4096
  float* eib = ws + 180224;    // 4096
  float* ejb = ws + 184320;    // 4096
  float* tb = ws + 188416;     // 65536  sim [16,4096]
  float* qca = ws + 253952;    // 131072
  float* kca = ws + 385024;    // 131072
  float* vca = ws + 516096;    // 131072
  float* scr = ws + 647168;    // 512
  float* att = ws + 647680;    // 512
  float* yca = ws + 648192;    // 131072
  float* t2 = ws + 779264;     // 65536
  float* fca = ws + 844800;    // 65536
  float* gca = ws + 910336;    // 131072
  float* t3 = ws + 1041408;    // 65536
  float* A1 = ws + 1106944;    // 1024
  float* B1 = ws + 1107968;    // 1024
  float* x1 = ws + 1108992;    // 65536
  float* A2 = ws + 1174528;    // 2048
  float* B2 = ws + 1176576;    // 2048
  float* x2 = ws + 1178624;    // 131072
  float* x3 = ws + 1309696;    // 4096
  float* x4 = ws + 1313792;    // 128
  float* f1o = ws + 1313920;   // 64
  float* f2o = ws + 1313984;   // 32
  float* f3o = ws + 1314016;   // 16
  float* part = ws + 1314048;  // 8*16*8192 = 1048576 (K-split partials)

  auto encode = [&](const float* x, const int* epair, int E, int n, float* e_out) {
    const int* src = epair;
    const int* dst = epair + E;
    k_gcn<<<1, 256, 0, stream>>>(x, P(P_gcn1_w), P(P_gcn1_b), src, dst, E, n, h1);
    k_gcn<<<1, 256, 0, stream>>>(h1, P(P_gcn2_w), P(P_gcn2_b), src, dst, E, n, h2);
    k_gcn<<<1, 256, 0, stream>>>(h2, P(P_gcn3_w), P(P_gcn3_b), src, dst, E, n, h3);
    k_add3_pad<<<16, 256, 0, stream>>>(h1, h2, h3, n, g);
    k_ln<<<64, 256, 0, stream>>>(g, P(P_san_g), P(P_san_b), yln, 64);
    k_gemm_rt<<<128, 256, 0, stream>>>(yln, P(P_q_w), P(P_q_b), nullptr, qb, 64, 64, 512, 0);
    k_gemm_rt<<<128, 256, 0, stream>>>(yln, P(P_k_w), P(P_k_b), nullptr, kb, 64, 64, 512, 0);
    k_gemm_rt<<<128, 256, 0, stream>>>(yln, P(P_v_w), P(P_v_b), nullptr, vb, 64, 64, 512, 0);
    k_attn_enc<<<512, 64, 0, stream>>>(qb, kb, vb, yb);
    k_gemm_rt<<<16, 256, 0, stream>>>(yb, P(P_o_w), P(P_o_b), g, sbuf, 64, 512, 64, 0);
    k_ln<<<64, 256, 0, stream>>>(sbuf, P(P_fn_g), P(P_fn_b), fbuf, 64);
    k_gemm_rt<<<64, 256, 0, stream>>>(fbuf, P(P_f1_w), P(P_f1_b), nullptr, ffn1, 64, 64, 256, 2);
    k_gemm_rt<<<16, 256, 0, stream>>>(ffn1, P(P_f2_w), P(P_f2_b), sbuf, s2, 64, 256, 64, 0);
    k_add<<<16, 256, 0, stream>>>(g, s2, e_out, 4096);  // return g + s
  };

  encode(xi, epi, E1, n1, eib);
  encode(xj, epj, E2, n2, ejb);

  // cross attention -> sim [16, 4096]
  k_gemm_rt<<<128, 256, 0, stream>>>(eib, P(P_cq_w), P(P_cq_b), nullptr, qb, 64, 64, 512, 0);
  k_gemm_rt<<<128, 256, 0, stream>>>(eib, P(P_ck_w), P(P_ck_b), nullptr, kb, 64, 64, 512, 0);
  k_gemm_rt<<<128, 256, 0, stream>>>(ejb, P(P_cq_w), P(P_cq_b), nullptr, vb, 64, 64, 512, 0);
  k_gemm_rt<<<128, 256, 0, stream>>>(ejb, P(P_ck_w), P(P_ck_b), nullptr, yb, 64, 64, 512, 0);
  k_cross<<<256, 256, 0, stream>>>(qb, kb, vb, yb, tb);

  // channel align: bandwidth-dominant WMMA GEMMs (805 MB of weights, K-split x8)
  const int KS = 8;
  auto big_gemm = [&](const float* Aact, const float* Wt, const float* bias,
                      const float* res, float* out, int K, int N, int act) {
    k_gemm_wmma_part<<<dim3(N / 128, KS), 128, 0, stream>>>(
        Aact, Wt, part, K, N, K / KS);
    k_gemm_finalize<<<(16 * N + 255) / 256, 256, 0, stream>>>(
        part, KS, bias, res, out, N, act);
  };
  big_gemm(tb, P(P_caq_w), P(P_caq_b), nullptr, qca, 4096, 8192, 0);
  big_gemm(tb, P(P_cak_w), P(P_cak_b), nullptr, kca, 4096, 8192, 0);
  big_gemm(tb, P(P_cav_w), P(P_cav_b), nullptr, vca, 4096, 8192, 0);
  k_ca_scores<<<512, 128, 0, stream>>>(qca, kca, scr);
  k_ca_softmax<<<1, 32, 0, stream>>>(scr, att);
  k_ca_av<<<512, 256, 0, stream>>>(att, vca, yca);
  big_gemm(yca, P(P_cao_w), P(P_cao_b), tb, t2, 8192, 4096, 0);
  k_ln<<<16, 256, 0, stream>>>(t2, P(P_cafn_g), P(P_cafn_b), fca, 4096);
  big_gemm(fca, P(P_caf1_w), P(P_caf1_b), nullptr, gca, 4096, 8192, 2);
  big_gemm(gca, P(P_caf2_w), P(P_caf2_b), t2, t3, 8192, 4096, 0);

  // sim CNN head
  k_e2e_proj<<<8, 128, 0, stream>>>(t3, P(P_e2e1a_w), P(P_e2e1a_b), A1, 16, 16, 0, 0);
  k_e2e_proj<<<8, 128, 0, stream>>>(t3, P(P_e2e1b_w), P(P_e2e1b_b), B1, 16, 16, 1, 0);
  k_e2e_combine<<<256, 256, 0, stream>>>(A1, B1, x1, 16);
  k_e2e_proj<<<16, 128, 0, stream>>>(x1, P(P_e2e2a_w), P(P_e2e2a_b), A2, 16, 32, 0, 0);
  k_e2e_proj<<<16, 128, 0, stream>>>(x1, P(P_e2e2b_w), P(P_e2e2b_b), B2, 16, 32, 1, 0);
  k_e2e_combine<<<512, 256, 0, stream>>>(A2, B2, x2, 32);
  k_e2e_proj<<<32, 128, 0, stream>>>(x2, P(P_e2n_w), P(P_e2n_b), x3, 32, 64, 0, 4);
  k_gemm_rt<<<1, 256, 0, stream>>>(x3, P(P_n2g_w), P(P_n2g_b), nullptr, x4, 1, 4096, 128, 4);
  k_gemm_rt<<<1, 256, 0, stream>>>(x4, P(P_fc1_w), P(P_fc1_b), nullptr, f1o, 1, 128, 64, 1);
  k_gemm_rt<<<1, 256, 0, stream>>>(f1o, P(P_fc2_w), P(P_fc2_b), nullptr, f2o, 1, 64, 32, 1);
  k_gemm_rt<<<1, 256, 0, stream>>>(f2o, P(P_fc3_w), P(P_fc3_b), nullptr, f3o, 1, 32, 16, 1);
  k_gemm_rt<<<1, 256, 0, stream>>>(f3o, P(P_fc4_w), P(P_fc4_b), nullptr, (float*)d_out, 1, 16, 1, 3);
}